// Induced_I2S_43834436223215
// MI455X (gfx1250) — compile-verified
//
#include <hip/hip_runtime.h>
#include <hip/hip_bf16.h>
#include <math.h>

// ---------------------------------------------------------------------------
// CDNA5 (gfx1250) implementation. Matrix math via V_WMMA_F32_16X16X4_F32
// (wave32, A=16x4 f32 in 2 VGPRs, B=4x16 f32 in 2 VGPRs, C/D=16x16 f32 in 8).
// F / B-operand streaming uses GLOBAL_LOAD_ASYNC_TO_LDS_B128 (ASYNCcnt).
// ---------------------------------------------------------------------------

typedef float v2f __attribute__((ext_vector_type(2)));
typedef float v4f __attribute__((ext_vector_type(4)));
typedef float v8f __attribute__((ext_vector_type(8)));

#define LMAX   8
#define S2DIM  81
#define SO3DIM 969
#define F_IN   81
#define F_SO3  300
#define CCH    768
#define KS     7
#define BATCH  8
#define G_S2   48
#define G_ACT  4096
#define G_SO3  36864
#define K1     (CCH * KS * KS)      // 37632 (divisible by 256)
#define N1     (F_IN * S2DIM)       // 6561

#if defined(__has_builtin)
#  if __has_builtin(__builtin_amdgcn_global_load_async_to_lds_b128)
#    define HAVE_ASYNC_LDS 1
#  endif
#endif

#ifdef HAVE_ASYNC_LDS
// Exact parameter types per hipcc diagnostic:
//   arg0: __attribute__((vector_size(16))) int  __device__ *   (AS1, non-const)
//   arg1: LDS-side pointer, same pointee        (AS3)
typedef int v4i_g __attribute__((vector_size(4 * sizeof(int))));
typedef __attribute__((address_space(1))) v4i_g as1_v4i;
typedef __attribute__((address_space(3))) v4i_g as3_v4i;

__device__ __forceinline__ void async_copy16(const float* gsrc, float* ldst) {
  __builtin_amdgcn_global_load_async_to_lds_b128(
      (as1_v4i*)(void*)gsrc, (as3_v4i*)(void*)ldst, 0, 0);
}
__device__ __forceinline__ void async_wait0() {
#  if __has_builtin(__builtin_amdgcn_s_wait_asynccnt)
  __builtin_amdgcn_s_wait_asynccnt(0);
#  else
  asm volatile("s_wait_asynccnt 0" ::: "memory");
#  endif
}
#endif

__constant__ int c_S2_OFF[10]  = {0, 1, 4, 9, 16, 25, 36, 49, 64, 81};
__constant__ int c_SO3_OFF[10] = {0, 1, 10, 35, 84, 165, 286, 455, 680, 969};

__device__ __forceinline__ v8f wmma4(v2f a, v2f b, v8f c) {
  // 8 args: (neg_a, A, neg_b, B, c_mod, C, reuse_a, reuse_b)
  return __builtin_amdgcn_wmma_f32_16x16x4_f32(false, a, false, b, (short)0, c,
                                               false, false);
}

// ---------------------------------------------------------------------------
// Kernel 1: y[b, j*81+i] = sum_{khw} x[b,khw] * F[(j*81+i), khw]
// F is a (6561 x 37632) row-major matrix -> NT skinny GEMM, BW-bound on F.
// LDS stride 260 dwords: multiple of 4 (16B-aligned b128 stores) and
// stride%64==4 (conflict-free fragment reads across 16 lanes).
// ---------------------------------------------------------------------------
__global__ __launch_bounds__(256) void k1_lift(const float* __restrict__ x,
                                               const float* __restrict__ Fm,
                                               float* __restrict__ y) {
  __shared__ float Xs[16][260];    // rows 8..15 stay zero (M padding)
  __shared__ float Fs[16][260];    // B operand, n-major
  __shared__ float Red[8][256];    // cross-wave reduction buffer

  const int t    = threadIdx.x;
  const int lane = t & 31;
  const int wv   = t >> 5;
  const int nt0  = blockIdx.x * 16;
  const bool fullN = (nt0 + 16 <= N1);

  // zero the padded A rows once
#pragma unroll
  for (int it = 0; it < 9; ++it) {
    int q = t + it * 256;                       // 8*260 = 2080 pad elements
    if (q < 8 * 260) Xs[8 + q / 260][q % 260] = 0.0f;
  }

  v8f acc = {};
  for (int k0 = 0; k0 < K1; k0 += 256) {
    // stage x: 8 rows x 256 -> 512 float4, 2 per thread, coalesced b128
#pragma unroll
    for (int it = 0; it < 2; ++it) {
      int q = t + (it << 8);
      int m = q >> 6, c = (q & 63) << 2;
      *(v4f*)&Xs[m][c] = *(const v4f*)&x[m * K1 + k0 + c];
    }
    // stage F: 16 rows x 256 -> 1024 float4, 4 per thread
#pragma unroll
    for (int it = 0; it < 4; ++it) {
      int q = t + (it << 8);
      int n = q >> 6, c = (q & 63) << 2;
      int row = nt0 + n;
      if (fullN || row < N1) {
        const float* src = &Fm[(size_t)row * K1 + (k0 + c)];
#ifdef HAVE_ASYNC_LDS
        async_copy16(src, &Fs[n][c]);
#else
        *(v4f*)&Fs[n][c] = *(const v4f*)src;
        if (it == 0 && k0 + 256 < K1)           // global_prefetch_b8
          __builtin_prefetch(&Fm[(size_t)row * K1 + (k0 + 256 + c)], 0, 1);
#endif
      } else {
        v4f zz = {0.0f, 0.0f, 0.0f, 0.0f};
        *(v4f*)&Fs[n][c] = zz;
      }
    }
#ifdef HAVE_ASYNC_LDS
    async_wait0();
#endif
    __syncthreads();

    const int kb   = wv << 5;                   // wave-private 32-wide K slice
    const int mrow = lane & 15;
    const int kh   = lane >> 4;
    const int ncol = lane & 15;
#pragma unroll
    for (int kk = 0; kk < 32; kk += 4) {
      int ks = kb + kk;
      v2f a, b;
      a.x = Xs[mrow][ks + 2 * kh];              // A: VGPR0 = K even (+2 per half)
      a.y = Xs[mrow][ks + 2 * kh + 1];          // A: VGPR1 = K odd
      b.x = Fs[ncol][ks + kh];                  // B: K rows striped across halves
      b.y = Fs[ncol][ks + 2 + kh];
      acc = wmma4(a, b, acc);
    }
    __syncthreads();
  }

  // reduce the 8 wave-private accumulators
#pragma unroll
  for (int r = 0; r < 8; ++r) Red[wv][r * 32 + lane] = acc[r];
  __syncthreads();

  const int r = t >> 5;                         // C VGPR index
  float s = 0.0f;
#pragma unroll
  for (int w = 0; w < 8; ++w) s += Red[w][r * 32 + lane];
  // C layout: lanes 0-15 -> M=r (the 8 real batches), N=lane; lanes 16-31 pad
  if (lane < 16) {
    int n = nt0 + lane;
    if (n < N1) y[r * N1 + n] = s;
  }
}

// ---------------------------------------------------------------------------
// Kernel 2: psi[f,g,u] = sum_n Y[n,u] * w_s2[f,g,n] / sqrt(48)
// ---------------------------------------------------------------------------
__global__ void k2_psi(const float* __restrict__ Y, const float* __restrict__ w_s2,
                       float* __restrict__ psi) {
  int idx = blockIdx.x * blockDim.x + threadIdx.x;
  if (idx >= F_IN * F_SO3 * S2DIM) return;
  int u = idx % S2DIM;
  int fg = idx / S2DIM;
  const float* w = w_s2 + fg * G_S2;
  float acc = 0.0f;
#pragma unroll 8
  for (int n = 0; n < G_S2; ++n) acc += Y[n * S2DIM + u] * w[n];
  psi[idx] = acc * 0.14433756729740643f;  // 1/sqrt(48)
}

// ---------------------------------------------------------------------------
// Kernel 3: z[b,g, SO3_OFF[l]+u*d+m] = (1/9) sum_f y[b,f,S2_OFF[l]+m]
//                                              * psi[f,g,S2_OFF[l]+u]
// ---------------------------------------------------------------------------
__global__ void k3_mix(const float* __restrict__ y, const float* __restrict__ psi,
                       float* __restrict__ z) {
  int idx = blockIdx.x * blockDim.x + threadIdx.x;
  if (idx >= BATCH * F_SO3 * SO3DIM) return;
  int i  = idx % SO3DIM;
  int bg = idx / SO3DIM;
  int g  = bg % F_SO3;
  int b  = bg / F_SO3;
  int l = 0;
  while (i >= c_SO3_OFF[l + 1]) ++l;
  int d = 2 * l + 1;
  int r = i - c_SO3_OFF[l];
  int u = r / d, m = r % d;
  int s2 = c_S2_OFF[l];
  const float* yp = y + b * N1 + s2 + m;                 // stride 81 over f
  const float* pp = psi + g * S2DIM + s2 + u;            // stride 300*81 over f
  float acc = 0.0f;
#pragma unroll 3
  for (int f = 0; f < F_IN; ++f) acc += yp[f * S2DIM] * pp[f * (F_SO3 * S2DIM)];
  z[idx] = acc * (1.0f / 9.0f);                          // 1/sqrt(F_IN)
}

// ---------------------------------------------------------------------------
// Generic WMMA f32 GEMM: C(MxN) = alpha * op(A(MxK) * B), optional relu store.
// TRANSB=1: B is (N x K) row-major (NT);  TRANSB=0: B is (K x N) row-major (NN).
// Block = 256 threads = 8 waves; tile 16(M) x 128(N); KC=64 LDS chunks.
// LDS stride 68 dwords -> stride%64==4, conflict-free fragment reads.
// ---------------------------------------------------------------------------
template <bool TRANSB, bool RELU>
__global__ __launch_bounds__(256) void gemm_wmma_f32(
    const float* __restrict__ A, const float* __restrict__ B,
    float* __restrict__ C, int M, int N, int K, float alpha) {
  constexpr int KC = 64;
  __shared__ float As[16][68];
  __shared__ float Bs[128][68];   // n-major

  const int t    = threadIdx.x;
  const int lane = t & 31;
  const int wv   = t >> 5;
  const int m0   = blockIdx.y * 16;
  const int n0   = blockIdx.x * 128;
  const bool fullM = (m0 + 16 <= M);
  const bool fullN = (n0 + 128 <= N);

  v8f acc = {};
  for (int k0 = 0; k0 < K; k0 += KC) {
    const bool fullK = (k0 + KC <= K);
    // ---- stage A: 16 x 64 = 4 elements/thread (coalesced along k) ----
    if (fullM && fullK) {
#pragma unroll
      for (int it = 0; it < 4; ++it) {
        int idx = t + (it << 8);
        int m = idx >> 6, k = idx & 63;
        As[m][k] = A[(size_t)(m0 + m) * K + (k0 + k)];
      }
    } else {
#pragma unroll
      for (int it = 0; it < 4; ++it) {
        int idx = t + (it << 8);
        int m = idx >> 6, k = idx & 63;
        int mg = m0 + m, kg = k0 + k;
        As[m][k] = (mg < M && kg < K) ? A[(size_t)mg * K + kg] : 0.0f;
      }
    }
    // ---- stage B: 128 x 64 = 32 elements/thread ----
    if (TRANSB) {  // B rows N-indexed, coalesced along k, stride-1 LDS writes
      if (fullN && fullK) {
#pragma unroll
        for (int it = 0; it < 32; ++it) {
          int idx = t + (it << 8);
          int n = idx >> 6, k = idx & 63;
          Bs[n][k] = B[(size_t)(n0 + n) * K + (k0 + k)];
        }
      } else {
#pragma unroll
        for (int it = 0; it < 32; ++it) {
          int idx = t + (it << 8);
          int n = idx >> 6, k = idx & 63;
          int ng = n0 + n, kg = k0 + k;
          Bs[n][k] = (ng < N && kg < K) ? B[(size_t)ng * K + kg] : 0.0f;
        }
      }
    } else {       // B rows K-indexed, coalesced along n
      if (fullN && fullK) {
#pragma unroll
        for (int it = 0; it < 32; ++it) {
          int idx = t + (it << 8);
          int k = idx >> 7, n = idx & 127;
          Bs[n][k] = B[(size_t)(k0 + k) * N + (n0 + n)];
        }
      } else {
#pragma unroll
        for (int it = 0; it < 32; ++it) {
          int idx = t + (it << 8);
          int k = idx >> 7, n = idx & 127;
          int ng = n0 + n, kg = k0 + k;
          Bs[n][k] = (ng < N && kg < K) ? B[(size_t)kg * N + ng] : 0.0f;
        }
      }
    }
    __syncthreads();

    const int mrow = lane & 15;
    const int kh   = lane >> 4;
    const int ncol = (wv << 4) + (lane & 15);
#pragma unroll
    for (int kk = 0; kk < KC; kk += 4) {
      v2f a, b;
      a.x = As[mrow][kk + 2 * kh];
      a.y = As[mrow][kk + 2 * kh + 1];
      b.x = Bs[ncol][kk + kh];
      b.y = Bs[ncol][kk + 2 + kh];
      acc = wmma4(a, b, acc);
    }
    __syncthreads();
  }

  const int kh    = lane >> 4;
  const int nglob = n0 + (wv << 4) + (lane & 15);
  const int mbase = m0 + (kh ? 8 : 0);
#pragma unroll
  for (int r = 0; r < 8; ++r) {
    int mg = mbase + r;
    if (mg < M && nglob < N) {
      float v = acc[r] * alpha;
      if (RELU) v = fmaxf(v, 0.0f);
      C[(size_t)mg * N + nglob] = v;
    }
  }
}

// ---------------------------------------------------------------------------
// Kernel 7: out[b, SO3_OFF[l]+v*d+m] =
//   (1/sqrt(300*d)) sum_{f,u} z2[b,f,off+u*d+m] * psi2[f,off+v*d+u]
// ---------------------------------------------------------------------------
__global__ void k7_final(const float* __restrict__ z2,
                         const float* __restrict__ psi2,
                         float* __restrict__ out) {
  int idx = blockIdx.x * blockDim.x + threadIdx.x;
  if (idx >= BATCH * SO3DIM) return;
  int i = idx % SO3DIM;
  int b = idx / SO3DIM;
  int l = 0;
  while (i >= c_SO3_OFF[l + 1]) ++l;
  int d = 2 * l + 1;
  int r = i - c_SO3_OFF[l];
  int v = r / d, m = r % d;
  int off = c_SO3_OFF[l];
  float acc = 0.0f;
  for (int f = 0; f < F_SO3; ++f) {
    const float* zr = z2 + ((size_t)b * F_SO3 + f) * SO3DIM + off;
    const float* pr = psi2 + (size_t)f * SO3DIM + off;
    float s = 0.0f;
    for (int u = 0; u < d; ++u) s += zr[u * d + m] * pr[v * d + u];
    acc += s;
  }
  out[idx] = acc * rsqrtf(300.0f * (float)d);
}

// ---------------------------------------------------------------------------
extern "C" void kernel_launch(void* const* d_in, const int* in_sizes, int n_in,
                              void* d_out, int out_size, void* d_ws, size_t ws_size,
                              hipStream_t stream) {
  const float* x     = (const float*)d_in[0];  // (8,768,7,7)
  const float* Fm    = (const float*)d_in[1];  // (81, 81*768, 7,7)
  const float* w_s2  = (const float*)d_in[2];  // (81,300,48)
  const float* Yc    = (const float*)d_in[3];  // (48,81)
  const float* w_so3 = (const float*)d_in[4];  // (300,1,36864)
  const float* Dm    = (const float*)d_in[5];  // (36864,969)
  const float* D_act = (const float*)d_in[6];  // (4096,969)
  float* out = (float*)d_out;                  // (8,1,969)

  float* ws    = (float*)d_ws;
  float* y     = ws;                 // 8*6561      =    52,488
  float* psi   = y + 52488;          // 81*300*81   = 1,968,300
  float* z     = psi + 1968300;      // 8*300*969   = 2,325,600
  float* srelu = z + 2325600;        // 2400*4096   = 9,830,400
  float* z2    = srelu + 9830400;    // 8*300*969   = 2,325,600
  float* psi2  = z2 + 2325600;       // 300*969     =   290,700
  (void)in_sizes; (void)n_in; (void)out_size; (void)ws_size;

  // 1. y = x . F^T   (8 x 37632) @ (37632 x 6561)
  k1_lift<<<(N1 + 15) / 16, 256, 0, stream>>>(x, Fm, y);

  // 2. psi = (Y^T . w_s2) / sqrt(48)
  k2_psi<<<(F_IN * F_SO3 * S2DIM + 255) / 256, 256, 0, stream>>>(Yc, w_s2, psi);

  // 3. z = per-l outer mix / sqrt(81)
  k3_mix<<<(BATCH * F_SO3 * SO3DIM + 255) / 256, 256, 0, stream>>>(y, psi, z);

  // 4. srelu = relu(z . D_act^T)       NT, (2400x969)@(969x4096)
  gemm_wmma_f32<true, true><<<dim3(32, 150), 256, 0, stream>>>(
      z, D_act, srelu, 2400, G_ACT, SO3DIM, 1.0f);

  // 5. z2 = srelu . D_act / 4096       NN, (2400x4096)@(4096x969)
  gemm_wmma_f32<false, false><<<dim3(8, 150), 256, 0, stream>>>(
      srelu, D_act, z2, 2400, SO3DIM, G_ACT, 1.0f / 4096.0f);

  // 6. psi2 = w_so3 . D / sqrt(36864)  NN, (300x36864)@(36864x969)
  gemm_wmma_f32<false, false><<<dim3(8, 19), 256, 0, stream>>>(
      w_so3, Dm, psi2, F_SO3, SO3DIM, G_SO3, 1.0f / 192.0f);

  // 7. final per-l contraction -> out
  k7_final<<<(BATCH * SO3DIM + 255) / 256, 256, 0, stream>>>(z2, psi2, out);
}